// Dynamic_MultiTeacher_26147760898281
// MI455X (gfx1250) — compile-verified
//
#include <hip/hip_runtime.h>
#include <hip/hip_bf16.h>
#include <stdint.h>

typedef __attribute__((ext_vector_type(2))) float v2f;
typedef __attribute__((ext_vector_type(8))) float v8f;
typedef __attribute__((ext_vector_type(4))) unsigned int u32x4;
typedef __attribute__((ext_vector_type(8))) int i32x8;
typedef __attribute__((ext_vector_type(4))) int i32x4;

#define ROWS 8
#define WSROW 18
#define TKD 20.0f
#define TTHR 6.0f

// ---------------------------------------------------------------------------
// TDM: DMA a (ROWS x ncols) fp32 tile from global memory into LDS.
// D# built per CDNA5 ISA ch.8: group0 = {count/lds_addr/global_addr/type},
// group1 = {data_size, tensor_dim0/1, tile_dim0/1, tensor_dim0_stride}.
// Groups 2/3 zero (2-D tensor). 6-arg clang-23 builtin form.
// ---------------------------------------------------------------------------
__device__ __forceinline__ void tdm_load_tile(unsigned lds_off, const float* gptr,
                                              int ncols, int nrows) {
  unsigned long long ga = (unsigned long long)(uintptr_t)gptr;
  u32x4 g0;
  g0[0] = 1u;                                                 // count=1 valid
  g0[1] = lds_off;                                            // lds_addr (bytes)
  g0[2] = (unsigned)ga;                                       // global_addr[31:0]
  g0[3] = ((unsigned)(ga >> 32) & 0x1FFFFFFu) | (2u << 30);   // addr[56:32] | type=2
  unsigned d0 = (unsigned)ncols;     // tensor_dim0 (elements)
  unsigned d1 = (unsigned)nrows;     // tensor_dim1 (rows available -> OOB rows read 0)
  i32x8 g1;
  g1[0] = (int)(2u << 16);                                    // data_size = 4 bytes
  g1[1] = (int)(d0 << 16);                                    // tensor_dim0[15:0]
  g1[2] = (int)((d0 >> 16) | (d1 << 16));                     // dim0 hi | dim1 lo
  g1[3] = (int)((d1 >> 16) | ((unsigned)ncols << 16));        // dim1 hi | tile_dim0
  g1[4] = ROWS;                                               // tile_dim1 | tile_dim2=0
  g1[5] = ncols;                                              // tensor_dim0_stride lo32
  g1[6] = 0;
  g1[7] = 0;
  i32x4 gz4 = {0, 0, 0, 0};
  i32x8 gz8 = {0, 0, 0, 0, 0, 0, 0, 0};
  __builtin_amdgcn_tensor_load_to_lds(g0, g1, gz4, gz4, gz8, 0);
}

// ---------------------------------------------------------------------------
// Phase 1: one wave per row. TDM stages 5 tensors x 8 rows in LDS; pass A
// computes top-2 / gathered / maxes, pass B the exp-sums; the 12 per-lane
// accumulators are cross-lane reduced with chained V_WMMA_F32_16X16X4_F32
// (B = ones => D[row q] = sum over 32 lanes of partial[q]).
// Per-row record in ws: [0..4]=margin [5..9]=gathered [10..14]=KD
//                       [15]=CE [16]=rowmax(4 teachers) [17]=rowmin(gathered4)
// ---------------------------------------------------------------------------
__global__ __launch_bounds__(256) void phase1_kernel(
    const float* __restrict__ t0g, const float* __restrict__ t1g,
    const float* __restrict__ t2g, const float* __restrict__ t3g,
    const float* __restrict__ sg,  const int* __restrict__ tgts,
    float* __restrict__ ws, int B, int C)
{
  extern __shared__ float dsm[];
  float* smem = dsm;                     // [5][ROWS][C]
  float* red  = dsm + 5 * ROWS * C;      // [ROWS][16][32]
  float* sums = red + ROWS * 16 * 32;    // [ROWS][16]

  const int tid  = threadIdx.x;
  const int wave = tid >> 5;
  const int lane = tid & 31;
  const int row0 = blockIdx.x * ROWS;

  if (wave == 0) {
    unsigned base = (unsigned)(uintptr_t)(void*)smem;   // LDS byte offset
    const float* gp[5] = { t0g, t1g, t2g, t3g, sg };
    int rem = B - row0;
    #pragma unroll
    for (int k = 0; k < 5; ++k)
      tdm_load_tile(base + (unsigned)(k * ROWS * C) * 4u,
                    gp[k] + (size_t)row0 * (size_t)C, C, rem);
    __builtin_amdgcn_s_wait_tensorcnt(0);
  }
  __syncthreads();

  const int  row   = row0 + wave;
  const bool valid = row < B;
  const float* L0 = smem + (0 * ROWS + wave) * C;
  const float* L1 = smem + (1 * ROWS + wave) * C;
  const float* L2 = smem + (2 * ROWS + wave) * C;
  const float* L3 = smem + (3 * ROWS + wave) * C;
  const float* LS = smem + (4 * ROWS + wave) * C;
  const int tgt = valid ? tgts[row] : 0;

  // ---- pass A: top-2 per branch (4 teachers + mimic), student max ----------
  float a1[5], a2[5];
  #pragma unroll
  for (int k = 0; k < 5; ++k) { a1[k] = -__builtin_inff(); a2[k] = -__builtin_inff(); }
  float ms = -__builtin_inff();
  for (int c = lane; c < C; c += 32) {
    float x0 = L0[c], x1 = L1[c], x2 = L2[c], x3 = L3[c], xs = LS[c];
    float xv[5] = { x0, x1, x2, x3, 0.25f * (x0 + x1 + x2 + x3) };
    #pragma unroll
    for (int k = 0; k < 5; ++k) {
      float x = xv[k];
      if (x > a1[k]) { a2[k] = a1[k]; a1[k] = x; }
      else           { a2[k] = fmaxf(a2[k], x); }
    }
    ms = fmaxf(ms, xs);
  }
  #pragma unroll
  for (int k = 0; k < 5; ++k) {
    float v1 = a1[k], v2 = a2[k];
    #pragma unroll
    for (int off = 16; off > 0; off >>= 1) {
      float o1 = __shfl_xor(v1, off);
      float o2 = __shfl_xor(v2, off);
      float n2 = fmaxf(fminf(v1, o1), fmaxf(v2, o2));
      v1 = fmaxf(v1, o1);
      v2 = n2;
    }
    a1[k] = v1; a2[k] = v2;
  }
  #pragma unroll
  for (int off = 16; off > 0; off >>= 1) ms = fmaxf(ms, __shfl_xor(ms, off));

  float g0v = L0[tgt], g1v = L1[tgt], g2v = L2[tgt], g3v = L3[tgt];
  float gth[5] = { g0v, g1v, g2v, g3v, 0.25f * (g0v + g1v + g2v + g3v) };
  float stgt = LS[tgt];

  // ---- pass B: exp-sums: Z_k, S_k = sum e*(t-s), student Z(T=20), Z(T=1) ---
  const float invT = 1.0f / TKD;
  float Z[5] = {0,0,0,0,0}, S[5] = {0,0,0,0,0}, Zs20 = 0.f, Zs1 = 0.f;
  for (int c = lane; c < C; c += 32) {
    float x0 = L0[c], x1 = L1[c], x2 = L2[c], x3 = L3[c], xs = LS[c];
    float xv[5] = { x0, x1, x2, x3, 0.25f * (x0 + x1 + x2 + x3) };
    #pragma unroll
    for (int k = 0; k < 5; ++k) {
      float e = __expf((xv[k] - a1[k]) * invT);
      Z[k] += e;
      S[k] += e * (xv[k] - xs);
    }
    Zs20 += __expf((xs - ms) * invT);
    Zs1  += __expf(xs - ms);
  }

  // ---- cross-lane sum of 12 accumulators via fp32 WMMA (B = ones) ----------
  float* rw = red + wave * (16 * 32);
  float part[12] = { Z[0],Z[1],Z[2],Z[3],Z[4], S[0],S[1],S[2],S[3],S[4], Zs20, Zs1 };
  #pragma unroll
  for (int q = 0; q < 12; ++q) rw[q * 32 + lane] = part[q];
  #pragma unroll
  for (int q = 12; q < 16; ++q) rw[q * 32 + lane] = 0.f;

  // A layout (16x4 f32): lane m<16 supplies A[m][0..1], lane m+16 supplies A[m][2..3]
  v8f acc = { 0.f,0.f,0.f,0.f,0.f,0.f,0.f,0.f };
  v2f bones; bones.x = 1.f; bones.y = 1.f;      // all-ones B (layout-independent)
  const int rsel = lane & 15;
  const int kb   = (lane < 16) ? 0 : 2;
  #pragma unroll
  for (int cc = 0; cc < 8; ++cc) {              // 8 chained K=4 chunks -> 32 lanes
    v2f a;
    a.x = rw[rsel * 32 + cc * 4 + kb];
    a.y = rw[rsel * 32 + cc * 4 + kb + 1];
    acc = __builtin_amdgcn_wmma_f32_16x16x4_f32(false, a, false, bones,
                                                (short)0, acc, false, false);
  }
  // D row m replicated over N: lane 0 VGPR j = sum[q=j], lane 16 VGPR j = sum[q=8+j]
  float* sw = sums + wave * 16;
  if (lane == 0) {
    #pragma unroll
    for (int j = 0; j < 8; ++j) sw[j] = acc[j];
  } else if (lane == 16) {
    #pragma unroll
    for (int j = 0; j < 4; ++j) sw[8 + j] = acc[j];
  }

  if (valid && lane == 0) {
    float LSEs = ms * invT + __logf(sw[10]);
    float* o = ws + (size_t)row * WSROW;
    #pragma unroll
    for (int k = 0; k < 5; ++k) {
      float Zk = sw[k], Sk = sw[5 + k];
      float margin = (a1[k] == gth[k]) ? (a1[k] - a2[k]) : 0.f;   // shift-invariant
      float LSEt = a1[k] * invT + __logf(Zk);
      float KD = TKD * (Sk / Zk) - (TKD * TKD) * (LSEt - LSEs);
      o[k]      = margin;
      o[5 + k]  = gth[k];
      o[10 + k] = KD;
    }
    o[15] = ms + __logf(sw[11]) - stgt;                            // CE
    o[16] = fmaxf(fmaxf(a1[0], a1[1]), fmaxf(a1[2], a1[3]));       // row max, teachers
    o[17] = fminf(fminf(gth[0], gth[1]), fminf(gth[2], gth[3]));   // row min, gathered
  }
}

// ---------------------------------------------------------------------------
// Phase 2: single block. Reduce global C (min gathered) and M (max logits),
// then fold the per-row weighted loss and write the scalar mean.
// ---------------------------------------------------------------------------
__global__ __launch_bounds__(256) void phase2_kernel(const float* __restrict__ ws,
                                                     float* __restrict__ out, int B)
{
  __shared__ float smn[256], smx[256], ssum[256];
  __shared__ float sh[2];
  const int tid = threadIdx.x;

  float mn = __builtin_inff(), mx = -__builtin_inff();
  for (int r = tid; r < B; r += 256) {
    const float* o = ws + (size_t)r * WSROW;
    mx = fmaxf(mx, o[16]);
    mn = fminf(mn, o[17]);
  }
  smn[tid] = mn; smx[tid] = mx;
  __syncthreads();
  for (int off = 128; off > 0; off >>= 1) {
    if (tid < off) {
      smn[tid] = fminf(smn[tid], smn[tid + off]);
      smx[tid] = fmaxf(smx[tid], smx[tid + off]);
    }
    __syncthreads();
  }
  if (tid == 0) {
    float Cm = smn[0];
    float shift = (Cm < 0.f) ? (-Cm + 1e-5f) : 0.f;
    sh[0] = shift;
    sh[1] = smx[0] + shift;      // max_preds
  }
  __syncthreads();
  const float shift = sh[0];
  const float invmaxp = 1.0f / sh[1];

  float acc = 0.f;
  for (int r = tid; r < B; r += 256) {
    const float* o = ws + (size_t)r * WSROW;
    float mrg[5], g[5], kd[5];
    #pragma unroll
    for (int k = 0; k < 5; ++k) { mrg[k] = o[k]; g[k] = o[5 + k]; kd[k] = o[10 + k]; }
    float CE = o[15];
    float pm = mrg[0];
    #pragma unroll
    for (int k = 1; k < 5; ++k) pm = fmaxf(pm, mrg[k]);
    float e[5], es = 0.f;
    #pragma unroll
    for (int k = 0; k < 5; ++k) { e[k] = __expf((mrg[k] - pm) * (1.0f / TTHR)); es += e[k]; }
    float inv_es = 1.0f / es;
    float loss = 0.f;
    #pragma unroll
    for (int k = 0; k < 5; ++k) {
      float w2 = (g[k] + shift) * invmaxp;
      float w1 = 1.f - w2;
      loss += (e[k] * inv_es) * (w1 * CE + w2 * kd[k]);
    }
    acc += loss;
  }
  ssum[tid] = acc;
  __syncthreads();
  for (int off = 128; off > 0; off >>= 1) {
    if (tid < off) ssum[tid] += ssum[tid + off];
    __syncthreads();
  }
  if (tid == 0) out[0] = ssum[0] / (float)B;
}

extern "C" void kernel_launch(void* const* d_in, const int* in_sizes, int n_in,
                              void* d_out, int out_size, void* d_ws, size_t ws_size,
                              hipStream_t stream) {
  (void)n_in; (void)out_size; (void)ws_size;
  const float* o1 = (const float*)d_in[0];
  const float* o2 = (const float*)d_in[1];
  const float* o3 = (const float*)d_in[2];
  const float* o4 = (const float*)d_in[3];
  const float* os = (const float*)d_in[4];
  const int*   tg = (const int*)d_in[5];
  int B = in_sizes[5];
  int C = in_sizes[0] / B;          // 8192 x 1000 in the reference
  float* ws  = (float*)d_ws;        // B * WSROW floats (~576 KB)
  float* out = (float*)d_out;

  int nblk = (B + ROWS - 1) / ROWS;
  size_t shmem = (size_t)(5 * ROWS * C + ROWS * 16 * 32 + ROWS * 16) * sizeof(float);
  phase1_kernel<<<nblk, 256, shmem, stream>>>(o1, o2, o3, o4, os, tg, ws, B, C);
  phase2_kernel<<<1, 256, 0, stream>>>(ws, out, B);
}